// GCN_42339787604897
// MI455X (gfx1250) — compile-verified
//
#include <hip/hip_runtime.h>

// ---------------------------------------------------------------------------
// 3-layer GCN for gfx1250 (MI455X).
//  - Dense transforms h = act(X) @ W via V_WMMA_F32_16X16X4_F32 (full f32).
//    W is staged in LDS pre-swizzled into B-fragment layout so each fragment
//    is one ds_load_b64; all 32 fragments of a column tile are preloaded so
//    the 32 WMMAs issue back-to-back on the matrix pipe.
//  - Edge aggregation via global_atomic_add_f32 (working set fits in 192MB L2).
// ---------------------------------------------------------------------------

typedef __attribute__((ext_vector_type(2))) float v2f;
typedef __attribute__((ext_vector_type(8))) float v8f;

#define KDIM 128           // input feature width of every layer in this network
#define NKK  (KDIM / 4)    // 32 K-groups of 4 per WMMA

// ---- degree / normalization ------------------------------------------------

__global__ void gcn_init_deg(float* __restrict__ deg, int n) {
  int i = blockIdx.x * blockDim.x + threadIdx.x;
  if (i < n) deg[i] = 1.0f;   // self-loop contribution
}

__global__ void gcn_accum_deg(const int* __restrict__ dst, float* __restrict__ deg, int E) {
  int e = blockIdx.x * blockDim.x + threadIdx.x;
  if (e < E) unsafeAtomicAdd(&deg[dst[e]], 1.0f);   // global_atomic_add_f32
}

__global__ void gcn_dinv(float* __restrict__ deg, int n) {
  int i = blockIdx.x * blockDim.x + threadIdx.x;
  if (i < n) {
    float d = deg[i];
    deg[i] = (d > 0.0f) ? rsqrtf(d) : 0.0f;   // deg >= 1 always, keep ref semantics
  }
}

// ---- dense transform: H = act(X) @ W  (X: n x 128, W: 128 x ncols) ---------
// One wave per 16-row strip. Fragment layouts per CDNA5 ISA 7.12.2:
//   A 16x4 f32 : lane m=l&15 ; v0 = A[m][2*(l>>4)]   , v1 = A[m][2*(l>>4)+1]
//   B 4x16 f32 : lane n=l&15 ; v0 = B[2*(l>>4)][n]   , v1 = B[2*(l>>4)+1][n]
//   C/D 16x16  : lane n=l&15 ; vgpr r -> M = r + 8*(l>>4)
// LDS holds W in fragment-major order: slot[(ct*NKK + kk)*32 + lane] = that
// lane's v2f for WMMA (ct, kk)  ->  one coalesced ds_load_b64 per fragment.
template <bool RELU>
__global__ void gcn_gemm_wmma(const float* __restrict__ X, const float* __restrict__ W,
                              float* __restrict__ H, int n, int ncols) {
  extern __shared__ v2f Wlds[];   // (ncols/16) * NKK * 32 fragments

  const int nct    = ncols >> 4;
  const int nslots = nct * NKK * 32;
  for (int idx = threadIdx.x; idx < nslots; idx += blockDim.x) {
    const int l   = idx & 31;
    const int fg  = idx >> 5;
    const int kk  = fg & (NKK - 1);        // NKK == 32, compile-time
    const int ct  = fg >> 5;               // fg / NKK
    const int mm  = l & 15;
    const int kbb = (l >> 4) << 1;
    const int k   = 4 * kk + kbb;
    const int col = (ct << 4) + mm;
    Wlds[idx] = (v2f){W[k * ncols + col], W[(k + 1) * ncols + col]};
  }
  __syncthreads();

  const int wave  = threadIdx.x >> 5;
  const int lane  = threadIdx.x & 31;
  const int strip = blockIdx.x * (blockDim.x >> 5) + wave;
  const int row0  = strip << 4;
  if (row0 >= n) return;               // wave-uniform: EXEC stays all-ones

  const int m  = lane & 15;
  const int hi = lane >> 4;            // 0 or 1
  const int kb = hi << 1;              // K sub-offset {0,2} inside a K=4 group

  // Load (and optionally ReLU) this wave's 16 x 128 A panel into registers.
  v2f afrag[NKK];
  const float* xrow = X + (size_t)(row0 + m) * KDIM;
#pragma unroll
  for (int kk = 0; kk < NKK; ++kk) {
    float a0 = xrow[4 * kk + kb + 0];
    float a1 = xrow[4 * kk + kb + 1];
    if (RELU) { a0 = fmaxf(a0, 0.0f); a1 = fmaxf(a1, 0.0f); }
    afrag[kk] = (v2f){a0, a1};
  }

  for (int ct = 0; ct < nct; ++ct) {
    // Preload all B fragments for this column tile (32 x ds_load_b64).
    v2f bfrag[NKK];
    const v2f* bp = Wlds + (size_t)ct * NKK * 32 + lane;
#pragma unroll
    for (int kk = 0; kk < NKK; ++kk) bfrag[kk] = bp[kk * 32];

    v8f acc = {};
#pragma unroll
    for (int kk = 0; kk < NKK; ++kk) {
      acc = __builtin_amdgcn_wmma_f32_16x16x4_f32(
          /*neg_a=*/false, afrag[kk], /*neg_b=*/false, bfrag[kk],
          /*c_mod=*/(short)0, acc, /*reuse_a=*/false, /*reuse_b=*/false);
    }
#pragma unroll
    for (int r = 0; r < 8; ++r) {
      const int row = row0 + r + (hi << 3);
      H[(size_t)row * ncols + ct * 16 + m] = acc[r];
    }
  }
}

// ---- out = bias + dinv[i]^2 * h   (self-loop term + bias, initializes out) -

__global__ void gcn_init_out(const float* __restrict__ h, const float* __restrict__ dinv,
                             const float* __restrict__ bias, float* __restrict__ out,
                             int total, int dshift, int dmask) {
  int t = blockIdx.x * blockDim.x + threadIdx.x;
  if (t >= total) return;
  const int i = t >> dshift;
  const int j = t & dmask;
  const float di = dinv[i];
  out[t] = bias[j] + di * di * h[t];
}

// ---- edge scatter: out[dst] += h[src] * dinv[src]*dinv[dst] ----------------
// One thread per (edge, float4-chunk). Working set fits in 192 MB L2, so the
// f32 atomics resolve at L2.

__global__ void gcn_scatter(const float* __restrict__ h, const int* __restrict__ src,
                            const int* __restrict__ dst, const float* __restrict__ dinv,
                            float* __restrict__ out, int E, int d,
                            int cshift, int cmask) {
  const unsigned t = blockIdx.x * blockDim.x + threadIdx.x;
  const unsigned e = t >> cshift;
  if (e >= (unsigned)E) return;
  const unsigned c = t & (unsigned)cmask;
  const int s  = src[e];
  const int dd = dst[e];
  const float nrm = dinv[s] * dinv[dd];
  const float4 v = ((const float4*)(h + (size_t)s * d))[c];
  float* op = out + (size_t)dd * d + 4u * c;
  unsafeAtomicAdd(op + 0, v.x * nrm);
  unsafeAtomicAdd(op + 1, v.y * nrm);
  unsafeAtomicAdd(op + 2, v.z * nrm);
  unsafeAtomicAdd(op + 3, v.w * nrm);
}

// ---------------------------------------------------------------------------

extern "C" void kernel_launch(void* const* d_in, const int* in_sizes, int n_in,
                              void* d_out, int out_size, void* d_ws, size_t ws_size,
                              hipStream_t stream) {
  (void)n_in; (void)out_size; (void)ws_size;

  const float* x  = (const float*)d_in[0];
  const int*   ei = (const int*)d_in[1];
  const float* W1 = (const float*)d_in[2];
  const float* b1 = (const float*)d_in[3];
  const float* W2 = (const float*)d_in[4];
  const float* b2 = (const float*)d_in[5];
  const float* W3 = (const float*)d_in[6];
  const float* b3 = (const float*)d_in[7];

  const int n    = in_sizes[0] / KDIM;   // 100000
  const int E    = in_sizes[1] / 2;      // 1600000
  const int dh   = in_sizes[3];          // 128
  const int dout = in_sizes[7];          // 64
  const int* src = ei;
  const int* dst = ei + E;

  float* wsf  = (float*)d_ws;
  size_t nPad = ((size_t)n + 255) & ~(size_t)255;
  float* dinv = wsf;                        // n floats
  float* bufA = wsf + nPad;                 // n * 128 floats (h buffer)
  float* bufB = bufA + (size_t)n * KDIM;    // n * 128 floats (agg buffer)

  const int TB = 256;

  // --- normalization coefficients (once; shared by all layers) ---
  gcn_init_deg<<<(n + TB - 1) / TB, TB, 0, stream>>>(dinv, n);
  gcn_accum_deg<<<(E + TB - 1) / TB, TB, 0, stream>>>(dst, dinv, E);
  gcn_dinv<<<(n + TB - 1) / TB, TB, 0, stream>>>(dinv, n);

  const int strips  = (n + 15) / 16;
  const int gblocks = (strips + 7) / 8;    // 8 waves (16-row strips) per block

  auto layer = [&](const float* in, const float* W, const float* b,
                   float* hbuf, float* obuf, int ncols, bool relu_in) {
    const size_t shmem = (size_t)KDIM * ncols * sizeof(float);  // same bytes, swizzled
    if (relu_in)
      gcn_gemm_wmma<true><<<gblocks, TB, shmem, stream>>>(in, W, hbuf, n, ncols);
    else
      gcn_gemm_wmma<false><<<gblocks, TB, shmem, stream>>>(in, W, hbuf, n, ncols);

    const int ds    = 31 - __builtin_clz((unsigned)ncols);  // log2(ncols)
    const int total = n * ncols;
    gcn_init_out<<<(total + TB - 1) / TB, TB, 0, stream>>>(hbuf, dinv, b, obuf,
                                                           total, ds, ncols - 1);
    const int cshift = ds - 2;                              // float4 chunks per row
    const unsigned stotal = (unsigned)E << cshift;
    gcn_scatter<<<(stotal + TB - 1) / TB, TB, 0, stream>>>(
        hbuf, src, dst, dinv, obuf, E, ncols, cshift, (1 << cshift) - 1);
  };

  // layer 1: conv(x) -> (relu fused into next gemm's load)
  layer(x, W1, b1, bufA, bufB, dh, /*relu_in=*/false);
  // layer 2: conv(relu(h1))
  layer(bufB, W2, b2, bufA, bufB, dh, /*relu_in=*/true);
  // layer 3: conv(relu(h2)) -> d_out (no final relu)
  layer(bufB, W3, b3, bufA, (float*)d_out, dout, /*relu_in=*/true);
}